// CnfGraphModel_15040975470805
// MI455X (gfx1250) — compile-verified
//
#include <hip/hip_runtime.h>
#include <hip/hip_bf16.h>

#define NVAR   50000
#define NRED   30000
#define NIRR   20000
#define NNODES 100000
#define NEDGES 1600000
#define HD     64
#define NEG_SLOPE 0.2f
#define TEMP   4.0f

typedef float v2f __attribute__((ext_vector_type(2)));
typedef float v8f __attribute__((ext_vector_type(8)));

__device__ __forceinline__ int node_t(int i) {
  return (i < NVAR) ? 0 : (i < NVAR + NRED ? 1 : 2);
}

// ---------------------------------------------------------------------------
// Embedding: h[i,k] = silu( [global(16) ; x(i,8)] @ Wfc + b ), per node class.
// Also zeroes the value accumulator (runs first in the launch sequence).
// ---------------------------------------------------------------------------
__global__ void embed_k(const float* __restrict__ x, const float* __restrict__ g,
                        const float* __restrict__ Wv, const float* __restrict__ bv,
                        const float* __restrict__ Wc, const float* __restrict__ bc,
                        float* __restrict__ h, float* __restrict__ vacc) {
  int tid = blockIdx.x * blockDim.x + threadIdx.x;
  if (tid == 0) vacc[0] = 0.f;
  if (tid >= NNODES * HD) return;
  int i = tid >> 6, k = tid & 63;
  const float* W; const float* b;
  if (i < NVAR) { W = Wv; b = bv; } else { W = Wc; b = bc; }
  float acc = b[k];
#pragma unroll
  for (int j = 0; j < 16; ++j) acc += g[j] * W[j * 64 + k];
  const float* xr = x + (size_t)i * 8;
#pragma unroll
  for (int j = 0; j < 8; ++j) acc += xr[j] * W[(16 + j) * 64 + k];
  h[tid] = acc / (1.f + expf(-acc));   // silu
}

// ---------------------------------------------------------------------------
// WMMA f32 GEMM: C(M x 64) = A(M x 64) @ B(64 x 64), M multiple of 16.
// One wave owns one 16-column slab (n_tile) and grid-strides over 16-row
// tiles; B slab (64x16) is preloaded into 32 VGPRs, K=64 done as 16x
// V_WMMA_F32_16X16X4_F32.
// RESID: C starts from Cin tile (pre-normalized agg), epilogue adds bias+ReLU.
// ---------------------------------------------------------------------------
template<bool RESID>
__global__ void gemm64_wmma_k(const float* __restrict__ A,
                              const float* __restrict__ B,
                              const float* __restrict__ Cin,
                              const float* __restrict__ bias,
                              float* __restrict__ Cout,
                              int mtiles) {
  const int lane   = threadIdx.x & 31;
  const int wave   = (blockIdx.x * blockDim.x + threadIdx.x) >> 5;
  const int nwaves = (gridDim.x * blockDim.x) >> 5;
  const int ntile  = wave & 3;              // which 16 of the 64 columns
  const int lane16 = lane & 15;
  const int khalf  = (lane >> 4) << 1;      // lanes 0-15 -> K+0/1, 16-31 -> K+2/3
  const int n      = ntile * 16 + lane16;

  // Preload B(64x16) slab: per-lane 16 x float2 covering K=0..63
  v2f breg[16];
#pragma unroll
  for (int kk = 0; kk < 16; ++kk) {
    int k = kk * 4 + khalf;
    breg[kk].x = B[k * 64 + n];
    breg[kk].y = B[(k + 1) * 64 + n];
  }

  for (int mt = (wave >> 2); mt < mtiles; mt += (nwaves >> 2)) {
    const int mbase = mt * 16 + ((lane < 16) ? 0 : 8);
    const float* arow = A + (size_t)(mt * 16 + lane16) * 64 + khalf;
    v8f acc;
    if (RESID) {
#pragma unroll
      for (int v = 0; v < 8; ++v) acc[v] = Cin[(size_t)(mbase + v) * 64 + n];
    } else {
      acc = (v8f){0.f, 0.f, 0.f, 0.f, 0.f, 0.f, 0.f, 0.f};
    }
#pragma unroll
    for (int kk = 0; kk < 16; ++kk) {
      v2f a = *(const v2f*)(arow + kk * 4);
      acc = __builtin_amdgcn_wmma_f32_16x16x4_f32(false, a, false, breg[kk],
                                                  (short)0, acc, false, false);
    }
#pragma unroll
    for (int v = 0; v < 8; ++v) {
      float val = acc[v];
      if (RESID) { val += bias[n]; val = val > 0.f ? val : 0.f; }
      Cout[(size_t)(mbase + v) * 64 + n] = val;
    }
  }
}

// ---------------------------------------------------------------------------
// Per-node attention scalars: asrc_t[i] = xt_t[i].att_src_t (all 3 types),
// adst[i] = xt_{type(i)}[i].att_dst_{type(i)} (own type only).
// ---------------------------------------------------------------------------
__global__ void attn_scalars_k(const float* __restrict__ xt3,
                               const float* as0, const float* as1, const float* as2,
                               const float* ad0, const float* ad1, const float* ad2,
                               float* __restrict__ asrc3, float* __restrict__ adst) {
  int i = blockIdx.x * blockDim.x + threadIdx.x;
  if (i >= NNODES) return;
  int ti = node_t(i);
  const float* asv[3] = {as0, as1, as2};
  const float* adv[3] = {ad0, ad1, ad2};
  for (int t = 0; t < 3; ++t) {
    const float* row = xt3 + ((size_t)t * NNODES + i) * HD;
    float s = 0.f;
#pragma unroll
    for (int k = 0; k < HD; ++k) s += row[k] * asv[t][k];
    asrc3[t * NNODES + i] = s;
    if (t == ti) {
      float sd = 0.f;
#pragma unroll
      for (int k = 0; k < HD; ++k) sd += row[k] * adv[t][k];
      adst[i] = sd;
    }
  }
}

// c_t = sum_k We_t[0,k] * att_edge_t[k]  (E_DIM == 1)
__global__ void edge_coef_k(const float* We0, const float* ae0,
                            const float* We1, const float* ae1,
                            const float* We2, const float* ae2,
                            float* __restrict__ c3) {
  int t = threadIdx.x;
  if (t >= 3) return;
  const float* We = t == 0 ? We0 : t == 1 ? We1 : We2;
  const float* ae = t == 0 ? ae0 : t == 1 ? ae1 : ae2;
  float s = 0.f;
  for (int k = 0; k < 64; ++k) s += We[k] * ae[k];
  c3[t] = s;
}

// agg=0, denom=0, m_enc = ordered-uint encoding of -inf
__global__ void clear_agg_k(float* __restrict__ agg, unsigned* __restrict__ menc,
                            float* __restrict__ denom) {
  int tid = blockIdx.x * blockDim.x + threadIdx.x;
  if (tid < NNODES * HD) agg[tid] = 0.f;
  if (tid < NNODES) { menc[tid] = 0x007FFFFFu; denom[tid] = 0.f; }
}

// Pass A: alpha per edge (dst-type attention), segment-max via uint atomicMax
__global__ void edge_alpha_max_k(const int* __restrict__ src, const int* __restrict__ dst,
                                 const float* __restrict__ eattr,
                                 const float* __restrict__ asrc3,
                                 const float* __restrict__ adst,
                                 const float* __restrict__ c3,
                                 float* __restrict__ alpha, unsigned* __restrict__ menc) {
  int e = blockIdx.x * blockDim.x + threadIdx.x;
  if (e >= NEDGES) return;
  int s = src[e], d = dst[e];
  int t = node_t(d);
  float a = asrc3[t * NNODES + s] + adst[d] + eattr[e] * c3[t];
  a = a > 0.f ? a : NEG_SLOPE * a;      // leaky_relu(., 0.2)
  alpha[e] = a;
  unsigned u = __float_as_uint(a);
  u = (u & 0x80000000u) ? ~u : (u | 0x80000000u);   // order-preserving encode
  atomicMax(menc + d, u);
}

// Pass B: decode max, non-finite -> 0 (matches jnp.where(isfinite(m), m, 0))
__global__ void decode_m_k(unsigned* __restrict__ menc) {
  int i = blockIdx.x * blockDim.x + threadIdx.x;
  if (i >= NNODES) return;
  unsigned u = menc[i];
  unsigned bits = (u & 0x80000000u) ? (u & 0x7FFFFFFFu) : ~u;
  float fv = __uint_as_float(bits);
  if ((bits & 0x7F800000u) == 0x7F800000u) fv = 0.f;
  ((float*)menc)[i] = fv;
}

// Pass C: one wave per edge. ea = exp(alpha - m[dst]); denom += ea;
// agg[dst] += xt_t[src] * ea  (normalized later, identical to per-edge w).
__global__ void edge_aggregate_k(const int* __restrict__ src, const int* __restrict__ dst,
                                 const float* __restrict__ alpha, const float* __restrict__ m,
                                 const float* __restrict__ xt3,
                                 float* __restrict__ denom, float* __restrict__ agg) {
  int gtid = blockIdx.x * blockDim.x + threadIdx.x;
  int e = gtid >> 5;
  int lane = threadIdx.x & 31;
  if (e >= NEDGES) return;
  int s = src[e], d = dst[e];
  int t = node_t(d);
  float ea = expf(alpha[e] - m[d]);
  if (lane == 0) atomicAdd(denom + d, ea);
  const float2 v = ((const float2*)(xt3 + ((size_t)t * NNODES + s) * HD))[lane];
  float* ag = agg + (size_t)d * HD + lane * 2;
  atomicAdd(ag,     v.x * ea);
  atomicAdd(ag + 1, v.y * ea);
}

__global__ void normalize_agg_k(float* __restrict__ agg, const float* __restrict__ denom) {
  int tid = blockIdx.x * blockDim.x + threadIdx.x;
  if (tid >= NNODES * HD) return;
  agg[tid] /= (denom[tid >> 6] + 1e-16f);
}

// logits (redundant clauses), pre-scaled by TEMPERATURE
__global__ void logits_k(const float* __restrict__ h, const float* __restrict__ wout,
                         float* __restrict__ logits) {
  int j = blockIdx.x * blockDim.x + threadIdx.x;
  if (j >= NRED) return;
  const float* row = h + (size_t)(NVAR + j) * HD;
  float s = 0.f;
#pragma unroll
  for (int k = 0; k < HD; ++k) s += row[k] * wout[k];
  logits[j] = s * TEMP;
}

// 100 Newton iterations for expected-value normalization; single workgroup,
// logits kept in registers, shuffle + LDS reductions.
__global__ void __launch_bounds__(1024) ev_norm_k(const float* __restrict__ logits,
                                                  const float* __restrict__ ex_ptr,
                                                  float* __restrict__ probs) {
  const int tid = threadIdx.x;
  float lg[30];
  int cnt = 0;
  for (int j = tid; j < NRED; j += 1024) lg[cnt++] = logits[j];
  __shared__ float s1s[32], s2s[32], bsh;
  float b = 0.f;
  const float ex = ex_ptr[0];
  for (int it = 0; it < 100; ++it) {
    float s1 = 0.f, s2 = 0.f;
    for (int c = 0; c < cnt; ++c) {
      float p = 1.f / (1.f + expf(-(lg[c] + b)));
      s1 += p; s2 += p * (1.f - p);
    }
    for (int off = 16; off > 0; off >>= 1) {
      s1 += __shfl_down(s1, off, 32);
      s2 += __shfl_down(s2, off, 32);
    }
    if ((tid & 31) == 0) { s1s[tid >> 5] = s1; s2s[tid >> 5] = s2; }
    __syncthreads();
    if (tid == 0) {
      float f = 0.f, fp = 0.f;
      for (int w = 0; w < 32; ++w) { f += s1s[w]; fp += s2s[w]; }
      f -= ex;
      float step = f / fp;
      step = fminf(2.f, fmaxf(-2.f, step));
      bsh = b - step;
    }
    __syncthreads();
    b = bsh;
  }
  cnt = 0;
  for (int j = tid; j < NRED; j += 1024)
    probs[j] = 1.f / (1.f + expf(-(lg[cnt++] + b)));
}

// value head: mean over all nodes of per-node linear head
__global__ void value_reduce_k(const float* __restrict__ h,
                               const float* __restrict__ wv, const float* __restrict__ bv,
                               const float* __restrict__ wc, float* __restrict__ vacc) {
  __shared__ float sh[256];
  int i = blockIdx.x * blockDim.x + threadIdx.x;
  float s = 0.f;
  if (i < NNODES) {
    const float* w = (i < NVAR) ? wv : wc;
    const float* row = h + (size_t)i * HD;
    float dotv = 0.f;
#pragma unroll
    for (int k = 0; k < HD; ++k) dotv += row[k] * w[k];
    if (i < NVAR) dotv += bv[0];
    s = dotv;
  }
  sh[threadIdx.x] = s;
  __syncthreads();
  for (int off = 128; off > 0; off >>= 1) {
    if (threadIdx.x < off) sh[threadIdx.x] += sh[threadIdx.x + off];
    __syncthreads();
  }
  if (threadIdx.x == 0) atomicAdd(vacc, sh[0]);
}

__global__ void value_final_k(const float* __restrict__ vacc, float* __restrict__ out) {
  out[0] = vacc[0] * (1.0f / (float)NNODES);
}

// ---------------------------------------------------------------------------
extern "C" void kernel_launch(void* const* d_in, const int* in_sizes, int n_in,
                              void* d_out, int out_size, void* d_ws, size_t ws_size,
                              hipStream_t stream) {
  (void)in_sizes; (void)n_in; (void)out_size; (void)ws_size;

  // Leaf layout (top-level insertion order; nested dicts flattened with
  // sorted keys, jax tree order):
  //  0..41 : blocks[b in 0..1][type in {irr,red,var}][W,We,Wres,att_dst,att_edge,att_src,bias]
  //  42/43 : clause_fc.W / .b     44: output_fc   45: value_clauses
  //  46/47 : value_vars.W / .b    48/49: variable_fc.W / .b
  //  50: x  51: edge_index  52: edge_attr  53: global_data  54: node_type
  //  55: ex 56: num_variables  57: num_redundant
  auto fp = [&](int i) { return (const float*)d_in[i]; };
  // node type t (0=var,1=red,2=irr) -> sorted param index P = 2 - t
  auto blkp = [&](int b, int t, int leaf) {
    return (const float*)d_in[b * 21 + (2 - t) * 7 + leaf];
  };
  enum { L_W = 0, L_We = 1, L_Wres = 2, L_att_dst = 3, L_att_edge = 4, L_att_src = 5, L_bias = 6 };

  const float* clause_W  = fp(42);
  const float* clause_b  = fp(43);
  const float* output_fc = fp(44);
  const float* value_cl  = fp(45);
  const float* value_vW  = fp(46);
  const float* value_vb  = fp(47);
  const float* var_W     = fp(48);
  const float* var_b     = fp(49);
  const float* x         = fp(50);
  const int*   ei        = (const int*)d_in[51];
  const float* eattr     = fp(52);
  const float* gdata     = fp(53);
  const float* ex_ptr    = fp(55);

  const int* src = ei;
  const int* dst = ei + NEDGES;

  // workspace carve-up (~160 MB)
  float* w = (float*)d_ws;
  float* h      = w;  w += (size_t)NNODES * HD;
  float* h2     = w;  w += (size_t)NNODES * HD;
  float* xt3    = w;  w += (size_t)3 * NNODES * HD;
  float* agg    = w;  w += (size_t)NNODES * HD;
  float* asrc3  = w;  w += (size_t)3 * NNODES;
  float* adst   = w;  w += NNODES;
  float* mbuf   = w;  w += NNODES;       // encoded max, then decoded float
  float* denom  = w;  w += NNODES;
  float* alpha  = w;  w += NEDGES;
  float* logits = w;  w += NRED;
  float* c3     = w;  w += 3;
  float* vacc   = w;  w += 1;

  const int T = 256;
  embed_k<<<(NNODES * HD + T - 1) / T, T, 0, stream>>>(
      x, gdata, var_W, var_b, clause_W, clause_b, h, vacc);

  const int start3[3] = {0, NVAR, NVAR + NRED};
  const int rows3[3]  = {NVAR, NRED, NIRR};

  for (int b = 0; b < 2; ++b) {
    const float* hin  = (b == 0) ? h : h2;
    float*       hout = (b == 0) ? h2 : h;

    for (int t = 0; t < 3; ++t)
      gemm64_wmma_k<false><<<128, T, 0, stream>>>(
          hin, blkp(b, t, L_W), nullptr, nullptr,
          xt3 + (size_t)t * NNODES * HD, NNODES / 16);

    edge_coef_k<<<1, 32, 0, stream>>>(blkp(b, 0, L_We), blkp(b, 0, L_att_edge),
                                      blkp(b, 1, L_We), blkp(b, 1, L_att_edge),
                                      blkp(b, 2, L_We), blkp(b, 2, L_att_edge), c3);

    attn_scalars_k<<<(NNODES + T - 1) / T, T, 0, stream>>>(
        xt3,
        blkp(b, 0, L_att_src), blkp(b, 1, L_att_src), blkp(b, 2, L_att_src),
        blkp(b, 0, L_att_dst), blkp(b, 1, L_att_dst), blkp(b, 2, L_att_dst),
        asrc3, adst);

    clear_agg_k<<<(NNODES * HD + T - 1) / T, T, 0, stream>>>(agg, (unsigned*)mbuf, denom);

    edge_alpha_max_k<<<(NEDGES + T - 1) / T, T, 0, stream>>>(
        src, dst, eattr, asrc3, adst, c3, alpha, (unsigned*)mbuf);

    decode_m_k<<<(NNODES + T - 1) / T, T, 0, stream>>>((unsigned*)mbuf);

    edge_aggregate_k<<<(NEDGES * 32 + T - 1) / T, T, 0, stream>>>(
        src, dst, alpha, mbuf, xt3, denom, agg);

    normalize_agg_k<<<(NNODES * HD + T - 1) / T, T, 0, stream>>>(agg, denom);

    for (int t = 0; t < 3; ++t)
      gemm64_wmma_k<true><<<128, T, 0, stream>>>(
          hin + (size_t)start3[t] * HD, blkp(b, t, L_Wres),
          agg + (size_t)start3[t] * HD, blkp(b, t, L_bias),
          hout + (size_t)start3[t] * HD, rows3[t] / 16);
  }

  // block 1 wrote back into h
  logits_k<<<(NRED + T - 1) / T, T, 0, stream>>>(h, output_fc, logits);
  ev_norm_k<<<1, 1024, 0, stream>>>(logits, ex_ptr, (float*)d_out);
  value_reduce_k<<<(NNODES + T - 1) / T, T, 0, stream>>>(h, value_vW, value_vb, value_cl, vacc);
  value_final_k<<<1, 1, 0, stream>>>(vacc, (float*)d_out + NRED);
}